// ModularQTS_73804718014525
// MI455X (gfx1250) — compile-verified
//
#include <hip/hip_runtime.h>
#include <math.h>

typedef float v2f __attribute__((ext_vector_type(2)));
typedef float v8f __attribute__((ext_vector_type(8)));

#define NQ    6
#define NT    8
#define DEG   4
#define NA    3
#define DQ    64
#define DA    8
#define NROTS 24
#define FDIM  64
#define BATCH 256

#define APAD 68   // LDS row stride (floats): breaks bank conflicts, keeps 8B align

// ============================================================================
// Kernel 1: projection GEMM + sigmoid using V_WMMA_F32_16X16X4_F32
//   ts[m][r] = sigmoid( sum_f x[b][f][t] * Wp[r][f] + bp[r] ),  m = b*8+t
//   M = 2048 (128 tiles of 16), N = 24 (2 tiles of 16, zero-padded), K = 64
//   Operands staged through LDS -> inner loop is ds_load_b64 + v_wmma only.
// ============================================================================
__global__ __launch_bounds__(32)
void proj_wmma_kernel(const float* __restrict__ x,
                      const float* __restrict__ Wp,
                      const float* __restrict__ bp,
                      float* __restrict__ ts)
{
    __shared__ float ldsA[16 * APAD];   // [row_local][k]  (transposed x tile)
    __shared__ float ldsW[32 * APAD];   // [n][k], rows 24..31 zeroed

    const int lane = threadIdx.x;       // 0..31 (wave32)

    // -- stage A: rows m = mtile*16 .. +15 are exactly 2 contiguous batches of x --
    const int b0 = blockIdx.x * 2;      // (mtile*16)/8
    for (int i = lane; i < 2 * FDIM * NT; i += 32) {   // 1024 coalesced floats
        const float v = x[b0 * (FDIM * NT) + i];
        const int bl = i >> 9, rem = i & 511;
        const int k = rem >> 3, t = rem & 7;
        ldsA[(bl * 8 + t) * APAD + k] = v;             // transpose: [row][k]
    }
    // -- stage W (rows 0..23), zero-pad rows 24..31; both loops wave-uniform --
    for (int i = lane; i < NROTS * FDIM; i += 32)      // 1536 = 48*32
        ldsW[(i >> 6) * APAD + (i & 63)] = Wp[i];
    for (int i = lane; i < 8 * APAD; i += 32)          // 544 = 17*32
        ldsW[NROTS * APAD + i] = 0.0f;
    __syncthreads();

    const int half = lane >> 4;         // lane-half selects K pair (0,1)/(2,3)
    const int l16  = lane & 15;
    const int n    = blockIdx.y * 16 + l16;            // output column (rot index)

    v8f acc = {};
    for (int kk = 0; kk < FDIM; kk += 4) {
        const int k0 = kk + 2 * half;
        // A: lane holds row l16; VGPR i -> K = 2*half + i
        const v2f a  = *(const v2f*)&ldsA[l16 * APAD + k0];
        // B: row K striped over lanes; VGPR i -> K = 2*half + i  (row n, padded)
        const v2f bb = *(const v2f*)&ldsW[(blockIdx.y * 16 + l16) * APAD + k0];
        acc = __builtin_amdgcn_wmma_f32_16x16x4_f32(false, a, false, bb,
                                                    (short)0, acc, false, false);
    }
    if (n < NROTS) {
        const float bv = bp[n];
        for (int v = 0; v < 8; ++v) {                  // C/D: VGPR v -> row v + 8*half
            const int row = blockIdx.x * 16 + v + 8 * half;
            const float z = acc[v] + bv;
            ts[row * NROTS + n] = 1.0f / (1.0f + expf(-z));
        }
    }
}

// ============================================================================
// Kernel 2: fused quantum circuit simulation, one workgroup per batch element
// ============================================================================
struct Gate { unsigned char ty, pa, pb, rot; };  // ty: 0=RY(bit pa), 1=CRX(ctrl pa, tgt pb)

// sim14 sequence (NQ=6, wire w <-> bit 5-w):
//  RY wires 0..5; CRX(ctrl=i,tgt=(i+1)%6) i=5..0; RY 0..5; CRX(ctrl=i,tgt=(i-1)%6) i=5,0,1,2,3,4
__device__ const Gate g_gates[24] = {
    {0,5,0, 0},{0,4,0, 1},{0,3,0, 2},{0,2,0, 3},{0,1,0, 4},{0,0,0, 5},
    {1,0,5, 6},{1,1,0, 7},{1,2,1, 8},{1,3,2, 9},{1,4,3,10},{1,5,4,11},
    {0,5,0,12},{0,4,0,13},{0,3,0,14},{0,2,0,15},{0,1,0,16},{0,0,0,17},
    {1,0,1,18},{1,5,0,19},{1,4,5,20},{1,3,4,21},{1,2,3,22},{1,1,2,23},
};

__device__ __forceinline__ int insert0(int v, int p) {
    return ((v >> p) << (p + 1)) | (v & ((1 << p) - 1));
}

// Apply the 24-gate sim14 sequence (or its adjoint) to all 8 ancilla columns.
// perT: angle tables indexed [a*NROTS + rot] (SELECT) vs [rot] (QFF, shared).
__device__ void applyGates(float* sRe, float* sIm,
                           const float* Ct, const float* St, bool perT, bool adj)
{
    const float sgn = adj ? -1.0f : 1.0f;
    for (int gi = 0; gi < 24; ++gi) {
        const Gate G = g_gates[adj ? 23 - gi : gi];
        if (G.ty == 0) {                                   // RY butterfly
            const int p = G.pa;
            for (int bid = threadIdx.x; bid < 256; bid += 64) {
                const int a = bid & 7, pr = bid >> 3;
                const int q0 = insert0(pr, p), q1 = q0 | (1 << p);
                const float c = perT ? Ct[a * NROTS + G.rot] : Ct[G.rot];
                const float s = (perT ? St[a * NROTS + G.rot] : St[G.rot]) * sgn;
                const int i0 = q0 * 8 + a, i1 = q1 * 8 + a;
                const float r0 = sRe[i0], m0 = sIm[i0], r1 = sRe[i1], m1 = sIm[i1];
                sRe[i0] = c * r0 - s * r1;  sIm[i0] = c * m0 - s * m1;
                sRe[i1] = s * r0 + c * r1;  sIm[i1] = s * m0 + c * m1;
            }
        } else {                                           // CRX: active only for ctrl=1
            const int pcb = G.pa, ptb = G.pb;
            const int lo = pcb < ptb ? pcb : ptb;
            const int hi = pcb < ptb ? ptb : pcb;
            for (int bid = threadIdx.x; bid < 128; bid += 64) {
                const int a = bid & 7, r4 = bid >> 3;
                const int q0 = insert0(insert0(r4, lo), hi) | (1 << pcb);
                const int q1 = q0 | (1 << ptb);
                const float c = perT ? Ct[a * NROTS + G.rot] : Ct[G.rot];
                const float s = (perT ? St[a * NROTS + G.rot] : St[G.rot]) * sgn;
                const int i0 = q0 * 8 + a, i1 = q1 * 8 + a;
                const float r0 = sRe[i0], m0 = sIm[i0], r1 = sRe[i1], m1 = sIm[i1];
                // [[c, -i*s],[-i*s, c]] on (z0,z1)
                sRe[i0] = c * r0 + s * m1;  sIm[i0] = c * m0 - s * r1;
                sRe[i1] = c * r1 + s * m0;  sIm[i1] = c * m1 - s * r0;
            }
        }
        __syncthreads();
    }
}

__global__ __launch_bounds__(64)
void qsim_kernel(const float* __restrict__ ts,       // (2048,24)
                 const float* __restrict__ prep,     // (3,3,2)
                 const float* __restrict__ sig,      // (5)
                 const float* __restrict__ qff,      // (24)
                 const float* __restrict__ A_off,    // (5,6)
                 const float* __restrict__ B_off,    // (5,6)
                 const float* __restrict__ D_diag,   // (5,4)
                 const float* __restrict__ W_head,   // (3,5)
                 const float* __restrict__ b_head,   // (3)
                 float* __restrict__ out)            // (256,3)
{
    __shared__ float sRe[DQ * DA], sIm[DQ * DA];
    __shared__ float angC[DA * NROTS], angS[DA * NROTS];
    __shared__ float qC[NROTS], qS[NROTS];
    __shared__ float VRe[64], VIm[64];
    __shared__ float HRe[5 * 16], HIm[5 * 16];
    __shared__ float part[5 * 64];
    __shared__ float exps[5];

    const int tid = threadIdx.x;
    const int b   = blockIdx.x;

    // -- per-(t,rot) cos/sin of ts/2 (reused across all DEG rounds) --
    for (int i = tid; i < DA * NROTS; i += 64) {
        const float th = 0.5f * ts[b * (NT * NROTS) + i];   // i = t*24 + r
        angC[i] = cosf(th);  angS[i] = sinf(th);
    }
    if (tid < NROTS) {
        const float th = 0.5f * qff[tid];
        qC[tid] = cosf(th);  qS[tid] = sinf(th);
    }

    // -- build PREPARE unitary V (8x8), one thread per column --
    if (tid < 8) {
        float vr[8], vi[8];
        for (int i = 0; i < 8; ++i) { vr[i] = 0.0f; vi[i] = 0.0f; }
        vr[tid] = 1.0f;
        for (int ly = 0; ly < 3; ++ly) {
            for (int q = 0; q < 3; ++q) {
                const int p = 2 - q;                       // wire q -> bit 2-q
                const float t0 = 0.5f * prep[(ly * 3 + q) * 2 + 0];
                const float c = cosf(t0), s = sinf(t0);    // RY
                for (int pr = 0; pr < 4; ++pr) {
                    const int r0 = insert0(pr, p), r1 = r0 | (1 << p);
                    const float a0 = vr[r0], b0 = vi[r0], a1 = vr[r1], b1 = vi[r1];
                    vr[r0] = c * a0 - s * a1;  vi[r0] = c * b0 - s * b1;
                    vr[r1] = s * a0 + c * a1;  vi[r1] = s * b0 + c * b1;
                }
                const float hz = 0.5f * prep[(ly * 3 + q) * 2 + 1];
                const float cz = cosf(hz), sz = sinf(hz);  // RZ: e^{-i hz}/e^{+i hz}
                for (int r = 0; r < 8; ++r) {
                    const float ph = ((r >> p) & 1) ? sz : -sz;
                    const float a0 = vr[r], b0 = vi[r];
                    vr[r] = a0 * cz - b0 * ph;  vi[r] = a0 * ph + b0 * cz;
                }
            }
            for (int i2 = 0; i2 < 2; ++i2) {               // CNOT(0,1), CNOT(1,2)
                const int pcb = 2 - i2, ptb = 1 - i2;
                for (int r = 0; r < 8; ++r)
                    if (((r >> pcb) & 1) && !((r >> ptb) & 1)) {
                        const int r1 = r | (1 << ptb);
                        const float tr = vr[r], ti = vi[r];
                        vr[r] = vr[r1]; vi[r] = vi[r1]; vr[r1] = tr; vi[r1] = ti;
                    }
            }
        }
        for (int i = 0; i < 8; ++i) { VRe[i * 8 + tid] = vr[i]; VIm[i * 8 + tid] = vi[i]; }
    }

    // -- build Hermitians H (4x4 each), one thread per observable --
    if (tid < 5) {
        const int rI[6] = {1, 2, 2, 3, 3, 3}, cI[6] = {0, 0, 1, 0, 1, 2};
        for (int i = 0; i < 16; ++i) { HRe[tid * 16 + i] = 0.0f; HIm[tid * 16 + i] = 0.0f; }
        for (int e = 0; e < 6; ++e) {
            const float ar = A_off[tid * 6 + e], br = B_off[tid * 6 + e];
            HRe[tid * 16 + rI[e] * 4 + cI[e]] = ar;  HIm[tid * 16 + rI[e] * 4 + cI[e]] = br;
            HRe[tid * 16 + cI[e] * 4 + rI[e]] = ar;  HIm[tid * 16 + cI[e] * 4 + rI[e]] = -br;
        }
        for (int d = 0; d < 3; ++d)
            HRe[tid * 16 + d * 4 + d] = 2.0f * D_diag[tid * 4 + d + 1];
    }

    // -- init |0...0> --
    for (int i = tid; i < DQ * DA; i += 64) {
        sRe[i] = (i == 0) ? 1.0f : 0.0f;
        sIm[i] = 0.0f;
    }
    __syncthreads();

    // projector-controlled phase: e^{+i phi} on anc==0, e^{-i phi} otherwise
    auto pcPhase = [&](float phi) {
        const float cp = cosf(phi), sp = sinf(phi);
        for (int i = tid; i < DQ * DA; i += 64) {
            const float sg = ((i & 7) == 0) ? sp : -sp;
            const float r0 = sRe[i], m0 = sIm[i];
            sRe[i] = r0 * cp - m0 * sg;
            sIm[i] = r0 * sg + m0 * cp;
        }
        __syncthreads();
    };

    // apply V (or V^dag) on the ancilla axis; thread q owns main-index q
    auto applyV = [&](bool dag) {
        float pr[8], pm[8], orr[8], oii[8];
        const int q = tid;
        for (int a = 0; a < 8; ++a) { pr[a] = sRe[q * 8 + a]; pm[a] = sIm[q * 8 + a]; }
        for (int i = 0; i < 8; ++i) {
            float ar = 0.0f, ai = 0.0f;
            for (int j = 0; j < 8; ++j) {
                const float vr = dag ? VRe[j * 8 + i] : VRe[i * 8 + j];
                const float vi = dag ? -VIm[j * 8 + i] : VIm[i * 8 + j];
                ar += vr * pr[j] - vi * pm[j];
                ai += vr * pm[j] + vi * pr[j];
            }
            orr[i] = ar;  oii[i] = ai;
        }
        for (int a = 0; a < 8; ++a) { sRe[q * 8 + a] = orr[a]; sIm[q * 8 + a] = oii[a]; }
        __syncthreads();
    };

    // -- QSP sequence --
    pcPhase(sig[0]);
    for (int k = 0; k < DEG; ++k) {
        applyV(false);
        applyGates(sRe, sIm, angC, angS, /*perT=*/true, /*adj=*/(k & 1) != 0); // SELECT
        applyV(true);
        pcPhase(sig[k + 1]);
    }
    applyGates(sRe, sIm, qC, qS, /*perT=*/false, /*adj=*/false);               // QFF

    // -- expectations of 5 sliding 2-local Hermitians --
    for (int w = 0; w < 5; ++w) {
        const int p0 = 4 - w;                 // wires (w,w+1) -> bits (p0+1, p0)
        float acc = 0.0f;
        for (int u = 0; u < 2; ++u) {
            const int m = tid * 2 + u;        // 0..127 residual columns
            const int a = m & 7, r4 = m >> 3;
            const int low = r4 & ((1 << p0) - 1), high = r4 >> p0;
            const int qb = (high << (p0 + 2)) | low;
            float vr[4], vi[4];
            for (int i = 0; i < 4; ++i) {
                const int idx = (qb | (i << p0)) * 8 + a;
                vr[i] = sRe[idx];  vi[i] = sIm[idx];
            }
            for (int i = 0; i < 4; ++i) {
                float zr = 0.0f, zi = 0.0f;
                for (int j = 0; j < 4; ++j) {
                    const float hr = HRe[w * 16 + i * 4 + j], hi = HIm[w * 16 + i * 4 + j];
                    zr += hr * vr[j] - hi * vi[j];
                    zi += hr * vi[j] + hi * vr[j];
                }
                acc += vr[i] * zr + vi[i] * zi;   // Re(conj(v_i) * (H v)_i)
            }
        }
        part[w * 64 + tid] = acc;
    }
    __syncthreads();
    if (tid < 5) {                            // fixed-order sum -> deterministic
        float s2 = 0.0f;
        for (int j = 0; j < 64; ++j) s2 += part[tid * 64 + j];
        exps[tid] = s2;
    }
    __syncthreads();
    if (tid < 3) {
        float o = b_head[tid];
        for (int w = 0; w < 5; ++w) o += exps[w] * W_head[tid * 5 + w];
        out[b * 3 + tid] = o;
    }
}

// ============================================================================
extern "C" void kernel_launch(void* const* d_in, const int* in_sizes, int n_in,
                              void* d_out, int out_size, void* d_ws, size_t ws_size,
                              hipStream_t stream)
{
    (void)in_sizes; (void)n_in; (void)out_size; (void)ws_size;
    const float* x      = (const float*)d_in[0];
    const float* Wp     = (const float*)d_in[1];
    const float* bp     = (const float*)d_in[2];
    const float* prep   = (const float*)d_in[3];
    const float* sig    = (const float*)d_in[4];
    const float* qff    = (const float*)d_in[5];
    const float* A_off  = (const float*)d_in[6];
    const float* B_off  = (const float*)d_in[7];
    const float* D_diag = (const float*)d_in[8];
    const float* W_head = (const float*)d_in[9];
    const float* b_head = (const float*)d_in[10];
    float* out = (float*)d_out;
    float* ts  = (float*)d_ws;     // 2048*24 floats = 192 KiB scratch

    proj_wmma_kernel<<<dim3(128, 2), 32, 0, stream>>>(x, Wp, bp, ts);
    qsim_kernel<<<dim3(BATCH), 64, 0, stream>>>(ts, prep, sig, qff, A_off, B_off,
                                                D_diag, W_head, b_head, out);
}